// OptNet_LearnD_41970420417257
// MI455X (gfx1250) — compile-verified
//
#include <hip/hip_runtime.h>
#include <math.h>

// OptNet PDIPM, MI455X (gfx1250, wave32).
// Structure: Q is diagonal -> 253x253 KKT solve reduces to a 64x64 SPD solve
// (q1*I + D^T diag(c) D) u = rhs.  H formed with V_WMMA_F32_16X16X4_F32,
// f32 Cholesky factor, f64 iterative refinement. One workgroup per batch elem.

#define NF 64
#define NM 63     // NF-1
#define NH 127    // 2*NF-1
#define NI 126    // 2*NF-2
#define BD 128    // 4 waves (wave32)
#define ITERS 30

typedef float v2f __attribute__((ext_vector_type(2)));
typedef float v8f __attribute__((ext_vector_type(8)));

__global__ __launch_bounds__(BD)
void optnet_pdipm_kernel(const float* __restrict__ x,
                         const float* __restrict__ Din,
                         float* __restrict__ out)
{
    const int tid = threadIdx.x;
    const int b   = blockIdx.x;
    const double q1  = 1.0 + 1.0e-4;        // Q diag, first 64
    const double q2  = 1.0e-8 + 1.0e-4;     // Q diag, last 63
    const double SIG = 0.1;

    __shared__ float  Dsh[64][64];          // D padded with zero row 63
    __shared__ float  Hs[64][66];           // H then its Cholesky factor (f32)
    __shared__ float  cf[64];               // f32 copy of c, cf[63]=0
    __shared__ double z[NH], lam[NI], s[NI];
    __shared__ double w[NI], r2[NI], dl[NI], dsv[NI];
    __shared__ double r1[NH];
    __shared__ double bv[NM], m22[NM], cc[NM], g2[NM], e[NM], y[NM], vv[NM];
    __shared__ double rhs[NF], uu[NF], resid[NF], delta[NF];
    __shared__ double red[BD];
    __shared__ double mu_s, alpha_s;

    // ---- load D (63x64) into LDS, pad row 63 with zeros ----
    for (int idx = tid; idx < 64 * 64; idx += BD) {
        int k = idx >> 6, j = idx & 63;
        Dsh[k][j] = (k < NM) ? Din[k * 64 + j] : 0.0f;
    }
    for (int j = tid; j < NH; j += BD) z[j] = 0.0;
    for (int j = tid; j < NI; j += BD) { lam[j] = 1.0; s[j] = 1.0; }
    if (tid == 0) cf[63] = 0.0f;
    __syncthreads();

    for (int it = 0; it < ITERS; ++it) {
        // ---- y = D * z[:64] ----
        if (tid < NM) {
            double acc = 0.0;
            for (int j = 0; j < NF; ++j) acc += (double)Dsh[tid][j] * z[j];
            y[tid] = acc;
        }
        // ---- mu = mean(s .* lam) ----
        red[tid] = (tid < NI) ? s[tid] * lam[tid] : 0.0;
        __syncthreads();
        if (tid == 0) {
            double t = 0.0;
            for (int j = 0; j < BD; ++j) t += red[j];
            mu_s = t / (double)NI;
        }
        // ---- r1 = -r_dual = -(Qz + p + G^T lam) ----
        if (tid < NF) {
            double acc = 0.0;
            for (int k = 0; k < NM; ++k)
                acc += (double)Dsh[k][tid] * (lam[k] - lam[NM + k]);
            double p = -(double)x[b * NF + tid];
            r1[tid] = -(q1 * z[tid] + p + acc);
        } else if (tid < NH) {
            int k = tid - NF;
            r1[tid] = -(q2 * z[tid] + 13.0 - (lam[k] + lam[NM + k]));
        }
        __syncthreads();
        const double mu = mu_s;
        // ---- r2 = -r_prim + s - SIG*mu/lam  (simplified) ----
        if (tid < NI) {
            int k = (tid < NM) ? tid : tid - NM;
            double base = (tid < NM) ? (-y[k] + z[NF + k]) : (y[k] + z[NF + k]);
            r2[tid] = base - SIG * mu / lam[tid];
            w[tid]  = lam[tid] / s[tid];
        }
        __syncthreads();
        // ---- Schur pieces: a, b, m22, c, g2, e ----
        if (tid < NM) {
            double w1 = w[tid], w2 = w[NM + tid];
            double a  = w1 + w2;
            double bb = w2 - w1;
            double m  = q2 + a;
            double c  = a - bb * bb / m;
            bv[tid] = bb; m22[tid] = m; cc[tid] = c; cf[tid] = (float)c;
            double gg2 = r1[NF + tid] - (w1 * r2[tid] + w2 * r2[NM + tid]);
            g2[tid] = gg2;
            e[tid]  = w1 * r2[tid] - w2 * r2[NM + tid] - bb * gg2 / m;
        }
        __syncthreads();
        if (tid < NF) {
            double acc = 0.0;
            for (int k = 0; k < NM; ++k) acc += (double)Dsh[k][tid] * e[k];
            rhs[tid] = r1[tid] + acc;
        }
        __syncthreads();

        // ---- H = D^T diag(c) D via V_WMMA_F32_16X16X4_F32 (all 4 waves) ----
        {
            const int wid  = tid >> 5;
            const int lane = tid & 31;
            const int mrow = lane & 15;
            const int half = lane >> 4;
            const int i0   = wid * 16;
            for (int tj = 0; tj < 4; ++tj) {
                const int j0 = tj * 16;
                v8f acc = {0.f, 0.f, 0.f, 0.f, 0.f, 0.f, 0.f, 0.f};
                for (int k0 = 0; k0 < 64; k0 += 4) {
                    int kb = k0 + half * 2;     // K=0,1 lanes 0-15; K=2,3 lanes 16-31
                    v2f a, bm;
                    a.x  = Dsh[kb][i0 + mrow];             // A = D^T tile
                    a.y  = Dsh[kb + 1][i0 + mrow];
                    bm.x = cf[kb]     * Dsh[kb][j0 + mrow];     // B = diag(c)*D tile
                    bm.y = cf[kb + 1] * Dsh[kb + 1][j0 + mrow];
                    acc = __builtin_amdgcn_wmma_f32_16x16x4_f32(
                              false, a, false, bm, (short)0, acc, false, false);
                }
                const int rbase = i0 + half * 8;   // C/D layout: rows r / r+8
                #pragma unroll
                for (int r = 0; r < 8; ++r)
                    Hs[rbase + r][j0 + mrow] = acc[r];
            }
        }
        __syncthreads();
        if (tid < NF) Hs[tid][tid] += (float)q1;
        __syncthreads();

        // ---- f32 Cholesky (lower), in place ----
        for (int k = 0; k < NF; ++k) {
            if (tid == 0) Hs[k][k] = sqrtf(Hs[k][k]);
            __syncthreads();
            float dk = Hs[k][k];
            for (int j = k + 1 + tid; j < NF; j += BD) Hs[j][k] /= dk;
            __syncthreads();
            int j = k + 1 + tid;
            if (j < NF) {
                float ljk = Hs[j][k];
                for (int i = k + 1; i <= j; ++i) Hs[j][i] -= ljk * Hs[i][k];
            }
            __syncthreads();
        }

        // ---- solve H u = rhs: f32 factor + f64 iterative refinement ----
        if (tid < NF) { uu[tid] = 0.0; resid[tid] = rhs[tid]; }
        __syncthreads();
        for (int pass = 0; pass < 3; ++pass) {
            if (tid < NF) delta[tid] = resid[tid];
            __syncthreads();
            for (int k = 0; k < NF; ++k) {            // forward L y = r
                if (tid == 0) delta[k] /= (double)Hs[k][k];
                __syncthreads();
                double dk = delta[k];
                int j = k + 1 + tid;
                if (j < NF) delta[j] -= (double)Hs[j][k] * dk;
                __syncthreads();
            }
            for (int k = NF - 1; k >= 0; --k) {       // backward L^T x = y
                if (tid == 0) delta[k] /= (double)Hs[k][k];
                __syncthreads();
                double dk = delta[k];
                if (tid < k) delta[tid] -= (double)Hs[k][tid] * dk;
                __syncthreads();
            }
            if (tid < NF) uu[tid] += delta[tid];
            __syncthreads();
            // exact f64 residual: resid = rhs - q1*u - D^T (c .* (D u))
            if (tid < NM) {
                double acc = 0.0;
                for (int j = 0; j < NF; ++j) acc += (double)Dsh[tid][j] * uu[j];
                y[tid] = acc;
                e[tid] = cc[tid] * acc;
            }
            __syncthreads();
            if (tid < NF) {
                double acc = 0.0;
                for (int k = 0; k < NM; ++k) acc += (double)Dsh[k][tid] * e[k];
                resid[tid] = rhs[tid] - q1 * uu[tid] - acc;
            }
            __syncthreads();
        }

        // ---- recover v, dlam, ds ----
        if (tid < NM) vv[tid] = (g2[tid] - bv[tid] * y[tid]) / m22[tid];
        __syncthreads();
        if (tid < NI) {
            int k = (tid < NM) ? tid : tid - NM;
            double sgn = (tid < NM) ? 1.0 : -1.0;
            double dli = w[tid] * (sgn * y[k] - vv[k] - r2[tid]);
            dl[tid]  = dli;
            dsv[tid] = -s[tid] + SIG * mu / lam[tid] - (s[tid] / lam[tid]) * dli;
        }
        __syncthreads();
        // ---- step length ----
        {
            double r = 1.0e300;
            if (tid < NI) {
                if (dl[tid]  < 0.0) r = fmin(r, -lam[tid] / dl[tid]);
                if (dsv[tid] < 0.0) r = fmin(r, -s[tid]  / dsv[tid]);
            }
            red[tid] = r;
        }
        __syncthreads();
        if (tid == 0) {
            double m = 1.0e300;
            for (int j = 0; j < BD; ++j) m = fmin(m, red[j]);
            alpha_s = fmin(1.0, 0.95 * m);
        }
        __syncthreads();
        double al = alpha_s;
        if (tid < NH) z[tid] += al * ((tid < NF) ? uu[tid] : vv[tid - NF]);
        if (tid < NI) { lam[tid] += al * dl[tid]; s[tid] += al * dsv[tid]; }
        __syncthreads();
    }

    if (tid < NF) out[b * NF + tid] = (float)z[tid];
}

extern "C" void kernel_launch(void* const* d_in, const int* in_sizes, int n_in,
                              void* d_out, int out_size, void* d_ws, size_t ws_size,
                              hipStream_t stream)
{
    const float* x = (const float*)d_in[0];   // (256, 64) f32
    const float* D = (const float*)d_in[1];   // (63, 64)  f32
    float* out = (float*)d_out;               // (256, 64) f32
    const int B = in_sizes[0] / NF;           // 256
    optnet_pdipm_kernel<<<B, BD, 0, stream>>>(x, D, out);
}